// general_attention_90400471646331
// MI455X (gfx1250) — compile-verified
//
#include <hip/hip_runtime.h>
#include <math.h>

#define B_   64
#define S_   4096
#define DX_  512
#define Q_   128
#define DQ_  512

#define CHUNK   256                 // S rows per workgroup
#define TILE    16                  // rows per LDS tile (32 KB per buffer)
#define NCHUNK  (S_ / CHUNK)        // 16 -> 1024 workgroups for the streaming pass
#define BLOCK   256                 // 8 wave32 per workgroup
#define PART_STRIDE (DX_ + 2)       // [m, l, ctx[512]] per (batch, chunk)

typedef float v2f __attribute__((ext_vector_type(2)));
typedef float v8f __attribute__((ext_vector_type(8)));

// ---------------------------------------------------------------------------
// CDNA5 async global->LDS copy (ASYNCcnt-tracked, bypasses VGPRs).
// VDST VGPR holds the LDS byte offset; low 32 bits of a flat shared-aperture
// pointer are exactly that offset on AMDGCN.
// ---------------------------------------------------------------------------
__device__ __forceinline__ void async_b128_to_lds(void* lds_ptr, const void* gptr) {
  unsigned lds_off = (unsigned)(uintptr_t)lds_ptr;
  unsigned long long ga = (unsigned long long)(uintptr_t)gptr;
  asm volatile("global_load_async_to_lds_b128 %0, %1, off"
               :: "v"(lds_off), "v"(ga) : "memory");
}
__device__ __forceinline__ void wait_async0() {
  asm volatile("s_wait_asynccnt 0x0" ::: "memory");
}

// ---------------------------------------------------------------------------
// Kernel 1: new_q[b,d] = sum_q p[q] * qm[b,q,d]   (reads 16 MB, BW bound)
// ---------------------------------------------------------------------------
__global__ void __launch_bounds__(256)
k_newq(const float* __restrict__ qm, const float* __restrict__ p,
       float* __restrict__ newq) {
  __shared__ float pl[Q_];
  const int b = blockIdx.x, tid = threadIdx.x;
  if (tid < Q_) pl[tid] = p[tid];
  __syncthreads();
  float a0 = 0.f, a1 = 0.f;
  for (int q = 0; q < Q_; ++q) {
    const float pv = pl[q];
    const float2 v = ((const float2*)(qm + ((size_t)b * Q_ + q) * DQ_))[tid];
    a0 = fmaf(pv, v.x, a0);
    a1 = fmaf(pv, v.y, a1);
  }
  newq[(size_t)b * DQ_ + 2 * tid]     = a0;
  newq[(size_t)b * DQ_ + 2 * tid + 1] = a1;
}

// ---------------------------------------------------------------------------
// Kernel 2: first = new_q @ W  via V_WMMA_F32_16X16X4_F32 (f32 precision kept).
// One wave per 16x16 output tile; K=512 in steps of 4.
// ---------------------------------------------------------------------------
__global__ void __launch_bounds__(32)
k_first_wmma(const float* __restrict__ newq, const float* __restrict__ W,
             float* __restrict__ first) {
  const int colTile = blockIdx.x & 31;   // 32 col tiles of 16
  const int rowTile = blockIdx.x >> 5;   // 4 row tiles of 16
  const int lane = threadIdx.x;
  const int half = lane >> 4;            // 0: K={k,k+1}, 1: K={k+2,k+3}
  const int l16  = lane & 15;
  const int rowBase = rowTile * 16;
  const int colBase = colTile * 16;

  v8f acc = {0.f, 0.f, 0.f, 0.f, 0.f, 0.f, 0.f, 0.f};
  const float* arow = newq + (size_t)(rowBase + l16) * DQ_ + 2 * half;

  for (int k = 0; k < DQ_; k += 4) {
    v2f a, bv;
    a.x = arow[k];
    a.y = arow[k + 1];
    const float* bp = W + (size_t)(k + 2 * half) * DX_ + colBase + l16;
    bv.x = bp[0];
    bv.y = bp[DX_];
    acc = __builtin_amdgcn_wmma_f32_16x16x4_f32(
        /*neg_a=*/false, a, /*neg_b=*/false, bv,
        /*c_mod=*/(short)0, acc, /*reuse_a=*/false, /*reuse_b=*/false);
  }
  #pragma unroll
  for (int r = 0; r < 8; ++r) {
    const int row = rowBase + r + 8 * half;
    first[(size_t)row * DX_ + colBase + l16] = acc[r];
  }
}

// ---------------------------------------------------------------------------
// Kernel 3: single-pass streaming softmax-attention over x (512 MB read once).
// Double-buffered LDS tiles filled by GLOBAL_LOAD_ASYNC_TO_LDS_B128; the
// async stream of tile t+1 overlaps VALU work on tile t.
// ---------------------------------------------------------------------------
__device__ __forceinline__ void stage_tile_async(const float* __restrict__ src,
                                                 float* dst, int tid) {
  #pragma unroll
  for (int i = 0; i < (TILE * DX_ / 4) / BLOCK; ++i) {   // 8 x b128 per thread
    const int e = tid + i * BLOCK;
    async_b128_to_lds((float4*)dst + e, (const float4*)src + e);
  }
}

__global__ void __launch_bounds__(BLOCK)
k_attn_partial(const float* __restrict__ x, const float* __restrict__ first,
               float* __restrict__ part) {
  extern __shared__ float smem[];
  float* xs0    = smem;                     // TILE*DX_ floats (32 KB)
  float* xs1    = xs0 + TILE * DX_;         // TILE*DX_ floats (32 KB)
  float* fl     = xs1 + TILE * DX_;         // DX_
  float* scores = fl + DX_;                 // TILE
  float* es     = scores + TILE;            // TILE
  float* mls    = es + TILE;                // [0]=m_run [1]=l_run [2]=scale

  const int tid  = threadIdx.x;
  const int wave = tid >> 5;
  const int lane = tid & 31;
  const int b = blockIdx.x / NCHUNK;
  const int c = blockIdx.x % NCHUNK;
  const size_t xbase = ((size_t)b * S_ + (size_t)c * CHUNK) * DX_;

  for (int i = tid; i < DX_; i += BLOCK) fl[i] = first[(size_t)b * DX_ + i];
  if (tid == 0) { mls[0] = -__builtin_inff(); mls[1] = 0.f; mls[2] = 0.f; }
  float cx0 = 0.f, cx1 = 0.f;               // each thread owns dims {2*tid, 2*tid+1}

  // Kick off tile 0 into buffer 0 (no barrier needed before issuing).
  stage_tile_async(x + xbase, xs0, tid);
  __syncthreads();                           // covers fl/mls init

  const int NT = CHUNK / TILE;               // 16 tiles
  for (int t = 0; t < NT; ++t) {
    float* xs = (t & 1) ? xs1 : xs0;
    // Tile t landed in LDS; also guarantees everyone finished reading the
    // buffer tile t+1 will overwrite (it was tile t-1's buffer).
    wait_async0();
    __syncthreads();
    if (t + 1 < NT)
      stage_tile_async(x + xbase + (size_t)(t + 1) * TILE * DX_,
                       (t & 1) ? xs0 : xs1, tid);

    // --- scores: one wave per row (8 waves, 16 rows -> 2 rows each) ---
    const float4* fl4 = (const float4*)fl;
    for (int r = wave; r < TILE; r += BLOCK / 32) {
      const float4* row4 = (const float4*)(xs + r * DX_);
      float acc = 0.f;
      #pragma unroll
      for (int j = 0; j < (DX_ / 4) / 32; ++j) {          // 4 float4 per lane
        const float4 xv = row4[lane + j * 32];
        const float4 fv = fl4[lane + j * 32];
        acc = fmaf(xv.x, fv.x, acc);
        acc = fmaf(xv.y, fv.y, acc);
        acc = fmaf(xv.z, fv.z, acc);
        acc = fmaf(xv.w, fv.w, acc);
      }
      #pragma unroll
      for (int off = 16; off >= 1; off >>= 1)
        acc += __shfl_xor(acc, off, 32);
      if (lane == 0) scores[r] = acc;
    }
    __syncthreads();

    // --- online softmax merge (wave 0) ---
    if (wave == 0) {
      float sc = (lane < TILE) ? scores[lane] : -__builtin_inff();
      float mt = sc;
      #pragma unroll
      for (int off = 16; off >= 1; off >>= 1)
        mt = fmaxf(mt, __shfl_xor(mt, off, 32));
      const float m_old = mls[0];
      const float m_new = fmaxf(m_old, mt);
      const float e  = __expf(sc - m_new);                // -inf lanes -> 0
      float lt = e;
      #pragma unroll
      for (int off = 16; off >= 1; off >>= 1)
        lt += __shfl_xor(lt, off, 32);
      if (lane < TILE) es[lane] = e;
      if (lane == 0) {
        const float scl = (m_old == -__builtin_inff()) ? 0.f : __expf(m_old - m_new);
        mls[2] = scl;
        mls[1] = mls[1] * scl + lt;
        mls[0] = m_new;
      }
    }
    __syncthreads();

    // --- context accumulate: ctx = ctx*scale + sum_r e[r]*x[r, 2*tid..] ---
    const float scl = mls[2];
    cx0 *= scl; cx1 *= scl;
    #pragma unroll
    for (int r = 0; r < TILE; ++r) {
      const float e = es[r];
      const float2 v = ((const float2*)(xs + r * DX_))[tid];
      cx0 = fmaf(e, v.x, cx0);
      cx1 = fmaf(e, v.y, cx1);
    }
    // Next iteration's top wait+barrier protects es/scores/mls and buffers.
  }

  float* pp = part + (size_t)(b * NCHUNK + c) * PART_STRIDE;
  if (tid == 0) { pp[0] = mls[0]; pp[1] = mls[1]; }
  pp[2 + 2 * tid]     = cx0;
  pp[2 + 2 * tid + 1] = cx1;
}

// ---------------------------------------------------------------------------
// Kernel 4: merge NCHUNK partials per batch with log-sum-exp rescale, divide.
// ---------------------------------------------------------------------------
__global__ void __launch_bounds__(256)
k_finalize(const float* __restrict__ part, float* __restrict__ out) {
  const int b = blockIdx.x, tid = threadIdx.x;
  const float* pb = part + (size_t)b * NCHUNK * PART_STRIDE;
  float M = -__builtin_inff();
  for (int ci = 0; ci < NCHUNK; ++ci) M = fmaxf(M, pb[(size_t)ci * PART_STRIDE]);
  float L = 0.f, c0 = 0.f, c1 = 0.f;
  for (int ci = 0; ci < NCHUNK; ++ci) {
    const float* pc = pb + (size_t)ci * PART_STRIDE;
    const float w = __expf(pc[0] - M);
    L += pc[1] * w;
    c0 = fmaf(w, pc[2 + 2 * tid], c0);
    c1 = fmaf(w, pc[2 + 2 * tid + 1], c1);
  }
  out[(size_t)b * DX_ + 2 * tid]     = c0 / L;
  out[(size_t)b * DX_ + 2 * tid + 1] = c1 / L;
}

// ---------------------------------------------------------------------------
extern "C" void kernel_launch(void* const* d_in, const int* in_sizes, int n_in,
                              void* d_out, int out_size, void* d_ws, size_t ws_size,
                              hipStream_t stream) {
  (void)in_sizes; (void)n_in; (void)out_size; (void)ws_size;
  const float* x  = (const float*)d_in[0];   // [B,S,DX]
  const float* qm = (const float*)d_in[1];   // [B,Q,DQ]
  const float* W  = (const float*)d_in[2];   // [DQ,DX]
  const float* p  = (const float*)d_in[3];   // [Q,1]
  float* out = (float*)d_out;                // [B,DX]

  float* ws    = (float*)d_ws;
  float* newq  = ws;                                   // B*DQ
  float* first = newq + (size_t)B_ * DQ_;              // B*DX
  float* part  = first + (size_t)B_ * DX_;             // B*NCHUNK*PART_STRIDE

  k_newq<<<B_, 256, 0, stream>>>(qm, p, newq);
  k_first_wmma<<<(B_ / 16) * (DX_ / 16), 32, 0, stream>>>(newq, W, first);

  const size_t smem =
      (size_t)(2 * TILE * DX_ + DX_ + TILE + TILE + 4) * sizeof(float);
  k_attn_partial<<<B_ * NCHUNK, BLOCK, smem, stream>>>(x, first, part);
  k_finalize<<<B_, 256, 0, stream>>>(part, out);
}